// TOADScaledDotProductAttentionLayer_87660282511606
// MI455X (gfx1250) — compile-verified
//
#include <hip/hip_runtime.h>
#include <math.h>

// ---------------------------------------------------------------------------
// TOAD scaled-dot-product attention (B=16, L=4096, H=1024, fp32)
//   sim[b,l] = <inputs[b,l,:], query[b,:]> / sqrt(2H)
//   att      = softmax_L(sim)
//   ctx[b,h] = sum_l att[b,l] * inputs[b,l,h]
// Memory-bound: 256 MB of inputs, ~11us at 23.3 TB/s for a single pass.
// Strategy: chunked online softmax. Kernel 1: per-(batch,chunk) block computes
// sims via V_WMMA_F32_16X16X4_F32, local (max, expsum), and unnormalized
// partial context via WMMA (chunk re-read hits L2). Kernel 2 merges chunks.
// ---------------------------------------------------------------------------

typedef __attribute__((ext_vector_type(2))) float v2f;
typedef __attribute__((ext_vector_type(4))) float v4f;
typedef __attribute__((ext_vector_type(8))) float v8f;

#define B_DIM 16
#define L_DIM 4096
#define H_DIM 1024
#define ROWS 128                  // rows per chunk = 8 waves * 16
#define NCHUNK (L_DIM / ROWS)     // 32
#define INV_SCALE 0.022097086912079608f   // 1/sqrt(2*1024)

__global__ __launch_bounds__(256)
void attn_partial_kernel(const float* __restrict__ inputs,
                         const float* __restrict__ query,
                         float* __restrict__ ws_hdr,   // [B][NCHUNK][2] = {max, expsum}
                         float* __restrict__ ws_ctx)   // [B][NCHUNK][H]
{
    const int chunk = blockIdx.x;
    const int b     = blockIdx.y;
    const int tid   = threadIdx.x;
    const int wave  = tid >> 5;
    const int lane  = tid & 31;
    const int nlow  = lane & 15;          // N index within 16-wide tile
    const int koff  = (lane >> 4) << 1;   // K offset: low half {0,1}, high half {2,3}

    __shared__ __align__(16) float lds_q[H_DIM];
    __shared__ __align__(16) float lds_sim[ROWS];
    __shared__ __align__(16) float lds_p[ROWS];

    // stage this batch's query row into LDS (4 KB)
    for (int i = tid; i < H_DIM; i += 256) lds_q[i] = query[b * H_DIM + i];
    __syncthreads();

    const size_t batch_off = (size_t)b * L_DIM * H_DIM;
    const int row0 = chunk * ROWS;

    // ---- phase 1: sims for 16 rows per wave via wmma f32 16x16x4 ----
    // A[m][k] = q[k0+k] (broadcast over m), B[k][n] = inputs[row0+w*16+n][k0+k]
    // => C[m][n] = sim[row0+w*16+n] replicated over m.
    {
        const int row = row0 + wave * 16 + nlow;
        const float* rp = inputs + batch_off + (size_t)row * H_DIM + koff;
        const float* qp = lds_q + koff;
        v8f acc = {};
        #pragma unroll 8
        for (int k0 = 0; k0 < H_DIM; k0 += 4) {
            v2f bv = *(const v2f*)(rp + k0);   // global_load_b64, lane-private row
            v2f av = *(const v2f*)(qp + k0);   // ds_load_b64, half-wave broadcast
            acc = __builtin_amdgcn_wmma_f32_16x16x4_f32(
                false, av, false, bv, (short)0, acc, false, false);
        }
        if (lane < 16) lds_sim[wave * 16 + lane] = acc[0] * INV_SCALE;
    }
    __syncthreads();

    // ---- local softmax stats over ROWS sims (LDS broadcast reads, cheap) ----
    float m = -3.0e38f;
    for (int i = 0; i < ROWS; ++i) m = fmaxf(m, lds_sim[i]);
    if (tid < ROWS) lds_p[tid] = __expf(lds_sim[tid] - m);
    __syncthreads();
    float s = 0.f;
    for (int i = 0; i < ROWS; ++i) s += lds_p[i];
    if (tid == 0) {
        ws_hdr[(b * NCHUNK + chunk) * 2 + 0] = m;
        ws_hdr[(b * NCHUNK + chunk) * 2 + 1] = s;
    }

    // ---- phase 2: unnormalized partial context via wmma ----
    // A[m][k] = p[l0+k] (broadcast over m), B[k][n] = inputs[row0+l0+k][h0+n]
    // => C[m][n] = sum_l p[l] * inputs[l][h0+n] replicated over m.
    const float* cbase = inputs + batch_off + (size_t)row0 * H_DIM;
    float* outc = ws_ctx + (size_t)(b * NCHUNK + chunk) * H_DIM;
    for (int t8 = 0; t8 < 8; ++t8) {
        const int h0 = wave * 128 + t8 * 16;
        v8f acc = {};
        #pragma unroll 4
        for (int l0 = 0; l0 < ROWS; l0 += 4) {
            v2f av = *(const v2f*)(lds_p + l0 + koff);        // LDS broadcast
            v2f bv;
            bv[0] = cbase[(size_t)(l0 + koff)     * H_DIM + h0 + nlow]; // coalesced (L2 hit)
            bv[1] = cbase[(size_t)(l0 + koff + 1) * H_DIM + h0 + nlow];
            acc = __builtin_amdgcn_wmma_f32_16x16x4_f32(
                false, av, false, bv, (short)0, acc, false, false);
        }
        if (lane < 16) outc[h0 + lane] = acc[0];
    }
}

__global__ __launch_bounds__(256)
void attn_combine_kernel(const float* __restrict__ ws_hdr,
                         const float* __restrict__ ws_ctx,
                         float* __restrict__ out)
{
    const int b   = blockIdx.x;
    const int tid = threadIdx.x;

    // global max + rescaled expsum total (redundant per thread, broadcast loads)
    float M = -3.0e38f;
    for (int c = 0; c < NCHUNK; ++c)
        M = fmaxf(M, ws_hdr[(b * NCHUNK + c) * 2]);
    float T = 0.f;
    for (int c = 0; c < NCHUNK; ++c)
        T += ws_hdr[(b * NCHUNK + c) * 2 + 1] * __expf(ws_hdr[(b * NCHUNK + c) * 2] - M);
    const float invT = 1.f / T;

    // each thread merges 4 h values
    const int h0 = tid * 4;
    float a0 = 0.f, a1 = 0.f, a2 = 0.f, a3 = 0.f;
    for (int c = 0; c < NCHUNK; ++c) {
        const float w = __expf(ws_hdr[(b * NCHUNK + c) * 2] - M);
        const v4f v = *(const v4f*)(ws_ctx + (size_t)(b * NCHUNK + c) * H_DIM + h0);
        a0 += w * v[0]; a1 += w * v[1]; a2 += w * v[2]; a3 += w * v[3];
    }
    v4f r; r[0] = a0 * invT; r[1] = a1 * invT; r[2] = a2 * invT; r[3] = a3 * invT;
    *(v4f*)(out + (size_t)b * H_DIM + h0) = r;
}

extern "C" void kernel_launch(void* const* d_in, const int* in_sizes, int n_in,
                              void* d_out, int out_size, void* d_ws, size_t ws_size,
                              hipStream_t stream) {
    const float* inputs = (const float*)d_in[0];   // [16, 4096, 1024] fp32
    const float* query  = (const float*)d_in[1];   // [16, 1024] fp32
    float* out = (float*)d_out;                    // [16, 1024] fp32

    float* hdr = (float*)d_ws;                         // 16*32*2 floats = 4 KB
    float* ctx = (float*)d_ws + B_DIM * NCHUNK * 2;    // 16*32*1024 floats = 2 MB

    dim3 grid1(NCHUNK, B_DIM);
    attn_partial_kernel<<<grid1, 256, 0, stream>>>(inputs, query, hdr, ctx);
    attn_combine_kernel<<<dim3(B_DIM), 256, 0, stream>>>(hdr, ctx, out);
}